// NoiseFilter_36094905155943
// MI455X (gfx1250) — compile-verified
//
#include <hip/hip_runtime.h>
#include <hip/hip_bf16.h>

#define BLOCK   256
#define ITEMS   16
#define TILE    (BLOCK * ITEMS)
#define WAVES   (BLOCK / 32)
#define MAXSEG  1024
#define MAX_RS  (MAXSEG + 1)
#define INFBITS 0x7F800000u
#define THRESH  0.1f

#if defined(__HIP_DEVICE_COMPILE__) && \
    __has_builtin(__builtin_amdgcn_global_load_async_to_lds_b128) && \
    __has_builtin(__builtin_amdgcn_s_wait_asynccnt)
#define USE_ASYNC_LDS 1
#endif

#ifdef USE_ASYNC_LDS
typedef int vint4 __attribute__((vector_size(16)));
typedef __attribute__((address_space(1))) vint4* gp4_t;   // global int4*
typedef __attribute__((address_space(3))) vint4* lp4_t;   // LDS int4*
#endif

// ---------------- device helpers ----------------

__device__ __forceinline__ int segsearch(const int* s_rs, int nseg, int i) {
    // largest seg with s_rs[seg] <= i
    int lo = 0, hi = nseg - 1;
    while (lo < hi) {
        int mid = (lo + hi + 1) >> 1;
        if (i >= s_rs[mid]) lo = mid; else hi = mid - 1;
    }
    return lo;
}

__device__ __forceinline__ void load16(const float* src, float* v) {
    const float4* p = (const float4*)src;   // 16B aligned by construction
#pragma unroll
    for (int q = 0; q < 4; ++q) {
        float4 t = p[q];
        v[4*q + 0] = t.x; v[4*q + 1] = t.y; v[4*q + 2] = t.z; v[4*q + 3] = t.w;
    }
}

// Stage one tile of `valid` floats from global g into LDS `lds` (16B-aligned).
// Caller must __syncthreads() after. Uses CDNA5 async global->LDS when available.
__device__ __forceinline__ void stage_tile(const float* g, float* lds, int valid, int tid) {
#ifdef USE_ASYNC_LDS
    if (valid == TILE) {
        const float* gp = g   + tid * ITEMS;
        float*       lp = lds + tid * ITEMS;
#pragma unroll
        for (int q = 0; q < 4; ++q) {
            __builtin_amdgcn_global_load_async_to_lds_b128(
                (gp4_t)(vint4*)(void*)(gp + 4*q),
                (lp4_t)(vint4*)(void*)(lp + 4*q), 0, 0);
        }
        __builtin_amdgcn_s_wait_asynccnt(0);
        return;
    }
#endif
    for (int k = tid; k < valid; k += BLOCK) lds[k] = g[k];
}

// Block-wide inclusive scan over one int per thread (wave32 shuffle + LDS combine).
// wsum must be __shared__ int[WAVES+1]. All threads must call. Returns inclusive sum,
// *total gets block total. Contains __syncthreads().
__device__ __forceinline__ int block_incl_scan(int v, int tid, int* wsum, int* total) {
    int lane = tid & 31, wid = tid >> 5;
    int incl = v;
#pragma unroll
    for (int d = 1; d < 32; d <<= 1) {
        int t = __shfl_up(incl, d, 32);
        if (lane >= d) incl += t;
    }
    if (lane == 31) wsum[wid] = incl;
    __syncthreads();
    if (tid == 0) {
        int acc = 0;
#pragma unroll
        for (int w = 0; w < WAVES; ++w) { int t = wsum[w]; wsum[w] = acc; acc += t; }
        wsum[WAVES] = acc;
    }
    __syncthreads();
    *total = wsum[WAVES];
    return incl + wsum[wid];
}

// ---------------- kernels ----------------

__global__ __launch_bounds__(BLOCK)
void k_init(unsigned* segmin, int* repPos, int* segCnt, int nseg, int n) {
    for (int k = threadIdx.x; k < nseg; k += BLOCK) {
        segmin[k] = INFBITS;
        repPos[k] = n;
        segCnt[k] = 0;
    }
}

__global__ __launch_bounds__(BLOCK)
void k_segmin(const float* __restrict__ score, const int* __restrict__ rs,
              int n, int nseg, unsigned* __restrict__ segmin) {
    __shared__ int s_rs[MAX_RS];
    __shared__ unsigned s_min[MAXSEG];
    const int tid = threadIdx.x;
    const int tileStart = blockIdx.x * TILE;
    __builtin_prefetch(score + tileStart, 0, 1);   // global_prefetch_b8
    for (int k = tid; k <= nseg; k += BLOCK) s_rs[k] = rs[k];
    for (int k = tid; k < nseg;  k += BLOCK) s_min[k] = INFBITS;
    __syncthreads();

    const int i0 = tileStart + tid * ITEMS;
    if (i0 < n) {
        const int cnt = min(ITEMS, n - i0);
        float v[ITEMS];
        if (cnt == ITEMS) load16(score + i0, v);
        else for (int j = 0; j < cnt; ++j) v[j] = score[i0 + j];

        int seg = segsearch(s_rs, nseg, i0);
        unsigned cur = INFBITS;
#pragma unroll
        for (int j = 0; j < ITEMS; ++j) {
            if (j >= cnt) break;
            int i = i0 + j;
            while (i >= s_rs[seg + 1]) {
                if (cur != INFBITS) atomicMin(&s_min[seg], cur);
                cur = INFBITS;
                ++seg;
            }
            unsigned b = __float_as_uint(v[j]);     // scores >= 0: bit order == float order
            cur = cur < b ? cur : b;
        }
        if (cur != INFBITS) atomicMin(&s_min[seg], cur);
    }
    __syncthreads();
    for (int k = tid; k < nseg; k += BLOCK)
        if (s_min[k] != INFBITS) atomicMin(&segmin[k], s_min[k]);
}

__global__ __launch_bounds__(BLOCK)
void k_count(const float* __restrict__ score, const int* __restrict__ rs,
             int n, int nseg, const unsigned* __restrict__ segmin,
             int* __restrict__ blkCnt, int* __restrict__ segCnt) {
    __shared__ int   s_rs[MAX_RS];
    __shared__ int   s_cnt[MAXSEG];
    __shared__ __align__(16) float s_tile[TILE];
    __shared__ int   wsum[WAVES + 1];
    const int tid = threadIdx.x;
    const int tileStart = blockIdx.x * TILE;
    const int valid = min(TILE, n - tileStart);
    for (int k = tid; k <= nseg; k += BLOCK) s_rs[k] = rs[k];
    for (int k = tid; k < nseg;  k += BLOCK) s_cnt[k] = 0;
    stage_tile(score + tileStart, s_tile, valid, tid);
    __syncthreads();

    const int i0 = tileStart + tid * ITEMS;
    int tot = 0;
    if (i0 < n) {
        const int cnt = min(ITEMS, n - i0);
        float v[ITEMS];
        if (cnt == ITEMS) load16(s_tile + tid * ITEMS, v);
        else for (int j = 0; j < cnt; ++j) v[j] = s_tile[tid * ITEMS + j];

        int seg = segsearch(s_rs, nseg, i0);
        unsigned smb = segmin[seg];
        int run = 0;
#pragma unroll
        for (int j = 0; j < ITEMS; ++j) {
            if (j >= cnt) break;
            int i = i0 + j;
            while (i >= s_rs[seg + 1]) {
                if (run) { atomicAdd(&s_cnt[seg], run); run = 0; }
                ++seg;
                smb = segmin[seg];
            }
            float x = v[j];
            int kp = (x > THRESH) || (x == __uint_as_float(smb));
            tot += kp; run += kp;
        }
        if (run) atomicAdd(&s_cnt[seg], run);
    }
    int total;
    (void)block_incl_scan(tot, tid, wsum, &total);
    if (tid == 0) blkCnt[blockIdx.x] = total;
    __syncthreads();
    for (int k = tid; k < nseg; k += BLOCK) {
        int c = s_cnt[k];
        if (c) atomicAdd(&segCnt[k], c);
    }
}

__global__ __launch_bounds__(BLOCK)
void k_scan(const int* __restrict__ blkCnt, int* __restrict__ blkOff, int nBlocks,
            const int* __restrict__ segCnt, int nseg, int* __restrict__ newrs) {
    __shared__ int wsum[WAVES + 1];
    const int tid = threadIdx.x;
    int running = 0;
    for (int c = 0; c < nBlocks; c += BLOCK) {
        int i = c + tid;
        int v = (i < nBlocks) ? blkCnt[i] : 0;
        int total;
        int incl = block_incl_scan(v, tid, wsum, &total);
        if (i < nBlocks) blkOff[i] = running + incl - v;
        running += total;
        __syncthreads();
    }
    if (tid == 0) {
        int acc = 0;
        newrs[0] = 0;
        for (int k = 0; k < nseg; ++k) { acc += segCnt[k]; newrs[k + 1] = acc; }
    }
}

__global__ __launch_bounds__(BLOCK)
void k_scatter(const float* __restrict__ score, const int* __restrict__ rs,
               int n, int nseg, const unsigned* __restrict__ segmin,
               const int* __restrict__ blkOff,
               int* __restrict__ sel, int* __restrict__ bg, int* __restrict__ repPos) {
    __shared__ int   s_rs[MAX_RS];
    __shared__ __align__(16) float s_tile[TILE];
    __shared__ int   wsum[WAVES + 1];
    const int tid = threadIdx.x;
    const int tileStart = blockIdx.x * TILE;
    const int valid = min(TILE, n - tileStart);
    for (int k = tid; k <= nseg; k += BLOCK) s_rs[k] = rs[k];
    stage_tile(score + tileStart, s_tile, valid, tid);
    __syncthreads();

    const int i0 = tileStart + tid * ITEMS;
    unsigned keepMask = 0, minMask = 0;
    int tot = 0, seg0 = 0, cnt = 0;
    float v[ITEMS];
    if (i0 < n) {
        cnt = min(ITEMS, n - i0);
        if (cnt == ITEMS) load16(s_tile + tid * ITEMS, v);
        else for (int j = 0; j < cnt; ++j) v[j] = s_tile[tid * ITEMS + j];

        int seg = segsearch(s_rs, nseg, i0);
        seg0 = seg;
        unsigned smb = segmin[seg];
#pragma unroll
        for (int j = 0; j < ITEMS; ++j) {
            if (j >= cnt) break;
            int i = i0 + j;
            while (i >= s_rs[seg + 1]) { ++seg; smb = segmin[seg]; }
            float x = v[j];
            unsigned ism = (x == __uint_as_float(smb)) ? 1u : 0u;
            unsigned kp  = ((x > THRESH) ? 1u : 0u) | ism;
            keepMask |= kp  << j;
            minMask  |= ism << j;
            tot += (int)kp;
        }
    }
    int total;
    int incl = block_incl_scan(tot, tid, wsum, &total);
    int base = blkOff[blockIdx.x] + incl - tot;   // exclusive global kept_pos base
    if (i0 < n) {
        int seg = seg0, rank = 0;
#pragma unroll
        for (int j = 0; j < ITEMS; ++j) {
            if (j >= cnt) break;
            int i = i0 + j;
            while (i >= s_rs[seg + 1]) ++seg;
            if ((keepMask >> j) & 1u) {
                int pos = base + rank; ++rank;
                bg[i] = pos;
                sel[pos] = i;
                if ((minMask >> j) & 1u) atomicMin(&repPos[seg], pos);
            }
        }
    }
}

__global__ __launch_bounds__(BLOCK)
void k_fix(const float* __restrict__ score, const int* __restrict__ rs,
           int n, int nseg, const unsigned* __restrict__ segmin,
           const int* __restrict__ repPos, int* __restrict__ bg) {
    __shared__ int s_rs[MAX_RS];
    const int tid = threadIdx.x;
    const int tileStart = blockIdx.x * TILE;
    __builtin_prefetch(score + tileStart, 0, 1);
    for (int k = tid; k <= nseg; k += BLOCK) s_rs[k] = rs[k];
    __syncthreads();

    const int i0 = tileStart + tid * ITEMS;
    if (i0 >= n) return;
    const int cnt = min(ITEMS, n - i0);
    float v[ITEMS];
    if (cnt == ITEMS) load16(score + i0, v);
    else for (int j = 0; j < cnt; ++j) v[j] = score[i0 + j];

    int seg = segsearch(s_rs, nseg, i0);
    unsigned smb = segmin[seg];
#pragma unroll
    for (int j = 0; j < ITEMS; ++j) {
        if (j >= cnt) break;
        int i = i0 + j;
        while (i >= s_rs[seg + 1]) { ++seg; smb = segmin[seg]; }
        float x = v[j];
        bool kp = (x > THRESH) || (x == __uint_as_float(smb));
        if (!kp) bg[i] = repPos[seg];
    }
}

// ---------------- host launcher ----------------

extern "C" void kernel_launch(void* const* d_in, const int* in_sizes, int n_in,
                              void* d_out, int out_size, void* d_ws, size_t ws_size,
                              hipStream_t stream) {
    const float* score = (const float*)d_in[0];
    const int*   rs    = (const int*)d_in[1];
    const int n    = in_sizes[0];       // total hits
    const int nrs  = in_sizes[1];       // nseg + 1
    const int nseg = nrs - 1;

    // d_out layout (int32, concatenated in reference return order):
    // sel[M], newrs[nseg+1], backgather[n]
    int* out   = (int*)d_out;
    const int M = out_size - n - nrs;
    int* sel   = out;
    int* newrs = out + M;
    int* bg    = out + M + nrs;

    const int nBlocks = (n + TILE - 1) / TILE;

    // workspace carve-up
    unsigned* segmin = (unsigned*)d_ws;
    int* repPos = (int*)(segmin + nseg);
    int* segCnt = repPos + nseg;
    int* blkCnt = segCnt + nseg;
    int* blkOff = blkCnt + nBlocks;
    (void)ws_size; (void)n_in; (void)M;

    k_init   <<<1,       BLOCK, 0, stream>>>(segmin, repPos, segCnt, nseg, n);
    k_segmin <<<nBlocks, BLOCK, 0, stream>>>(score, rs, n, nseg, segmin);
    k_count  <<<nBlocks, BLOCK, 0, stream>>>(score, rs, n, nseg, segmin, blkCnt, segCnt);
    k_scan   <<<1,       BLOCK, 0, stream>>>(blkCnt, blkOff, nBlocks, segCnt, nseg, newrs);
    k_scatter<<<nBlocks, BLOCK, 0, stream>>>(score, rs, n, nseg, segmin, blkOff, sel, bg, repPos);
    k_fix    <<<nBlocks, BLOCK, 0, stream>>>(score, rs, n, nseg, segmin, repPos, bg);
}